// KCAT_54090818126584
// MI455X (gfx1250) — compile-verified
//
#include <hip/hip_runtime.h>
#include <hip/hip_bf16.h>

typedef __attribute__((ext_vector_type(2))) float v2f;
typedef __attribute__((ext_vector_type(8))) float v8f;

#define EPS_BN   1e-5f
#define EPS_NORM 1e-7f

constexpr int B_ = 512, K_ = 2, N_ = 32;
constexpr int C1_ = 4, F1_ = 16;
constexpr int NCLS_ = 40;
constexpr int JROWS = N_ + 1;                 // 33 rows: [x ; 32 neighbors]
constexpr int HNB_STRIDE = JROWS * C1_ * F1_; // 33*64 = 2112

// fast HW ops (single v_sqrt_f32 / v_rcp_f32 / v_rsq_f32)
__device__ __forceinline__ float fsqrt(float a)  { return __builtin_amdgcn_sqrtf(a); }
__device__ __forceinline__ float frcp(float a)   { return __builtin_amdgcn_rcpf(a); }
__device__ __forceinline__ float frsq(float a)   { return __builtin_amdgcn_rsqf(a); }
__device__ __forceinline__ float sgnroot(float a){ return copysignf(fsqrt(fabsf(a)), a); }
__device__ __forceinline__ float softsign(float z){ return z * frcp(1.f + fabsf(z)); }

// ---------------------------------------------------------------------------
// Kernel 1: per (b,k) workgroup, 256 threads (8 wave32).
//   s[g]    = sum_n nb[n,g]
//   c[g]    = sum_f sqrt(|x[f]s[g]+x[g]s[f]|)   (column normalizer; A symmetric)
//   out[j,g]= (sum_f u[j,f] * sgnroot(x[f]s[g]+x[g]s[f])) / (c[g]+eps)   [WMMA]
//   hnb[j,col] = sum_g out[j,g] * W1k[g,col]                              [WMMA]
// ---------------------------------------------------------------------------
__global__ __launch_bounds__(256) void k1_graph(
    const float* __restrict__ x, const float* __restrict__ nb,
    const float* __restrict__ W1, float* __restrict__ hnb)
{
  extern __shared__ float smem[];
  float* u    = smem;                       // 33*512
  float* outm = smem + JROWS * 512;         // 33*512
  float* lx   = smem + 2 * JROWS * 512;     // 512
  float* ls   = lx + 512;                   // 512
  float* lic  = ls + 512;                   // 512

  const int tid = threadIdx.x;
  const int b = blockIdx.x >> 1;
  const int k = blockIdx.x & 1;
  const float* w1k = W1 + (size_t)k * 4 * 8192;   // (C1,F0,F1) -> 4*512*16

  // warm L2 with this hop's W1 slab (128KB, shared by all 512 batches)
  for (int i = tid; i < 512; i += 256)
    __builtin_prefetch(w1k + i * 64, 0, 1);

  // row 0 = x
  for (int i = tid; i < 512; i += 256) {
    float v = x[b * 512 + i];
    u[i] = v; lx[i] = v;
  }
  // rows 1..32 = neighbors (vectorized)
  const float4* nb4 = (const float4*)(nb + (size_t)(b * 2 + k) * N_ * 512);
  for (int i = tid; i < (N_ * 512) / 4; i += 256) {
    float4 v = nb4[i];
    int n = (i * 4) >> 9, f = (i * 4) & 511;
    float* dst = &u[(1 + n) * 512 + f];
    dst[0] = v.x; dst[1] = v.y; dst[2] = v.z; dst[3] = v.w;
  }
  __syncthreads();

  // s[g]
  for (int g = tid; g < 512; g += 256) {
    float acc = 0.f;
    for (int n = 1; n <= N_; ++n) acc += u[n * 512 + g];
    ls[g] = acc;
  }
  __syncthreads();

  // invc[g] = 1/(sum_f |sgnroot(.)| + eps)
  for (int g = tid; g < 512; g += 256) {
    float xg = lx[g], sg = ls[g], c = 0.f;
    for (int f = 0; f < 512; ++f)
      c += fsqrt(fabsf(lx[f] * sg + xg * ls[f]));
    lic[g] = frcp(c + EPS_NORM);
  }
  __syncthreads();

  const int lane = tid & 31, w = tid >> 5;
  const int hi = lane >> 4, lo = lane & 15;
  const v2f vzero = {0.f, 0.f};

  // ---- Stage A: out = U @ A ; 2 g-tiles per pass so each LDS frag feeds 6 wmma
  for (int pass = 0; pass < 2; ++pass) {
    const int gta = w + pass * 16;          // gtiles gta and gta+8
    const int ga = gta * 16 + lo, gb_ = (gta + 8) * 16 + lo;
    const float xga = lx[ga], sga = ls[ga], ica = lic[ga];
    const float xgb = lx[gb_], sgb = ls[gb_], icb = lic[gb_];
    v8f a00 = {}, a01 = {}, a02 = {};
    v8f a10 = {}, a11 = {}, a12 = {};
    for (int f0 = 0; f0 < 512; f0 += 4) {
      int fb = f0 + 2 * hi;                 // this lane's K pair: K = 2*hi + {0,1}
      v2f xp = *(const v2f*)&lx[fb];
      v2f sp = *(const v2f*)&ls[fb];
      v2f af0 = *(const v2f*)&u[lo * 512 + fb];
      v2f af1 = *(const v2f*)&u[(16 + lo) * 512 + fb];
      v2f t32 = *(const v2f*)&u[32 * 512 + fb];
      v2f af2 = (lo == 0) ? t32 : vzero;    // rows 33..47 are zero padding
      v2f bfa, bfb;
      bfa.x = sgnroot(xp.x * sga + xga * sp.x);
      bfa.y = sgnroot(xp.y * sga + xga * sp.y);
      bfb.x = sgnroot(xp.x * sgb + xgb * sp.x);
      bfb.y = sgnroot(xp.y * sgb + xgb * sp.y);
      a00 = __builtin_amdgcn_wmma_f32_16x16x4_f32(false, af0, false, bfa, (short)0, a00, false, false);
      a01 = __builtin_amdgcn_wmma_f32_16x16x4_f32(false, af1, false, bfa, (short)0, a01, false, false);
      a02 = __builtin_amdgcn_wmma_f32_16x16x4_f32(false, af2, false, bfa, (short)0, a02, false, false);
      a10 = __builtin_amdgcn_wmma_f32_16x16x4_f32(false, af0, false, bfb, (short)0, a10, false, false);
      a11 = __builtin_amdgcn_wmma_f32_16x16x4_f32(false, af1, false, bfb, (short)0, a11, false, false);
      a12 = __builtin_amdgcn_wmma_f32_16x16x4_f32(false, af2, false, bfb, (short)0, a12, false, false);
    }
#pragma unroll
    for (int i = 0; i < 8; ++i) {           // D: vgpr i -> M = i + 8*hi
      int r = hi * 8 + i;
      outm[r * 512 + ga]         = a00[i] * ica;
      outm[(16 + r) * 512 + ga]  = a01[i] * ica;
      outm[r * 512 + gb_]        = a10[i] * icb;
      outm[(16 + r) * 512 + gb_] = a11[i] * icb;
      if (hi == 0 && i == 0) {
        outm[32 * 512 + ga]  = a02[0] * ica;
        outm[32 * 512 + gb_] = a12[0] * icb;
      }
    }
  }
  __syncthreads();

  // ---- Stage B: hnb = out @ W1k  (48x64, K=512), W1k streamed from L2
  float* outp = hnb + (size_t)(b * 2 + k) * HNB_STRIDE;
  for (int pp = 0; pp < 2; ++pp) {
    int p = w + pp * 8;
    if (p >= 12) break;                     // 12 (mt,nt) tiles over 8 waves
    int mt = p % 3, nt = p / 3;
    int col = nt * 16 + lo;                 // col = d*16 + e
    const float* wcol = w1k + (col >> 4) * 8192 + (col & 15);
    v8f acc = {};
    for (int g0 = 0; g0 < 512; g0 += 4) {
      int gb = g0 + 2 * hi;
      v2f bf;
      bf.x = wcol[gb * 16];
      bf.y = wcol[gb * 16 + 16];
      v2f af;
      if (mt < 2) {
        af = *(const v2f*)&outm[(mt * 16 + lo) * 512 + gb];
      } else {
        v2f t = *(const v2f*)&outm[32 * 512 + gb];
        af = (lo == 0) ? t : vzero;
      }
      acc = __builtin_amdgcn_wmma_f32_16x16x4_f32(false, af, false, bf, (short)0, acc, false, false);
    }
#pragma unroll
    for (int i = 0; i < 8; ++i) {
      int row = mt * 16 + hi * 8 + i;
      if (row < JROWS) outp[row * 64 + col] = acc[i];
    }
  }
}

// ---------------------------------------------------------------------------
// Kernel 2: BN1 stats for h: per (k,d) over (b,e) -> 8 blocks
// ---------------------------------------------------------------------------
__global__ __launch_bounds__(256) void k2_bn1_stats(
    const float* __restrict__ hnb, float* __restrict__ st1)
{
  __shared__ float rs[256], rq[256];
  int kd = blockIdx.x, k = kd >> 2, d = kd & 3, tid = threadIdx.x;
  float s = 0.f, q = 0.f;
  for (int i = tid; i < 8192; i += 256) {
    int b = i >> 4, e = i & 15;
    float v = hnb[(size_t)(b * 2 + k) * HNB_STRIDE + d * 16 + e];
    s += v; q += v * v;
  }
  rs[tid] = s; rq[tid] = q; __syncthreads();
  for (int o = 128; o > 0; o >>= 1) {
    if (tid < o) { rs[tid] += rs[tid + o]; rq[tid] += rq[tid + o]; }
    __syncthreads();
  }
  if (tid == 0) {
    float m = rs[0] / 8192.f;
    st1[kd * 2] = m;
    st1[kd * 2 + 1] = rq[0] / 8192.f - m * m;
  }
}

// ---------------------------------------------------------------------------
// Kernel 3: per (b,k): BN+softsign, fadj2 (16x16), transform with W2 -> h2raw
// ---------------------------------------------------------------------------
__global__ __launch_bounds__(128) void k3_layer2(
    const float* __restrict__ hnb, const float* __restrict__ st1,
    const float* __restrict__ g1, const float* __restrict__ b1,
    const float* __restrict__ W2, float* __restrict__ h2raw)
{
  __shared__ float raw[JROWS * 64];
  __shared__ float hact[64], snb[64];
  __shared__ float scl[4], sft[4];
  __shared__ float fn[256], icn[16], xg2[64];
  __shared__ float red[4][32][2];

  int tid = threadIdx.x;
  int b = blockIdx.x >> 1, k = blockIdx.x & 1;
  const float* src = hnb + (size_t)(b * 2 + k) * HNB_STRIDE;
  for (int i = tid; i < JROWS * 64; i += 128) raw[i] = src[i];
  __syncthreads();

  if (tid < 64) {  // h: BN (cross-batch stats) + softsign
    int c = tid >> 4;
    float m = st1[(k * 4 + c) * 2], v = st1[(k * 4 + c) * 2 + 1];
    float z = (raw[tid] - m) * frsq(v + EPS_BN) * g1[k * 4 + c] + b1[k * 4 + c];
    hact[tid] = softsign(z);
  }
  {  // nbh per-(b,c) stats over (n,e): 512 values
    int c = tid >> 5, t2 = tid & 31;
    float s = 0.f, q = 0.f;
    const float* rp = &raw[(1 + t2) * 64 + c * 16];
    for (int e = 0; e < 16; ++e) { float v = rp[e]; s += v; q += v * v; }
    red[c][t2][0] = s; red[c][t2][1] = q;
  }
  __syncthreads();
  if (tid < 4) {
    float s = 0.f, q = 0.f;
    for (int i = 0; i < 32; ++i) { s += red[tid][i][0]; q += red[tid][i][1]; }
    float m = s / 512.f, var = q / 512.f - m * m;
    float sc = frsq(var + EPS_BN) * g1[k * 4 + tid];
    scl[tid] = sc; sft[tid] = b1[k * 4 + tid] - m * sc;
  }
  __syncthreads();
  if (tid < 64) {  // snb[c,g] = sum_n softsign(bn(nbh))
    int c = tid >> 4, g = tid & 15;
    float acc = 0.f, sc = scl[c], sh = sft[c];
    for (int n = 1; n <= N_; ++n)
      acc += softsign(raw[n * 64 + c * 16 + g] * sc + sh);
    snb[tid] = acc;
  }
  __syncthreads();
  for (int idx = tid; idx < 256; idx += 128) {  // symmetrized adjacency + sgnroot
    int f = idx >> 4, g = idx & 15;
    float t = 0.f;
    for (int c = 0; c < 4; ++c)
      t += hact[c * 16 + f] * snb[c * 16 + g] + hact[c * 16 + g] * snb[c * 16 + f];
    fn[idx] = sgnroot(t);
  }
  __syncthreads();
  if (tid < 16) {  // column normalizer (axis-1 sum)
    float cn = 0.f;
    for (int f = 0; f < 16; ++f) cn += fabsf(fn[f * 16 + tid]);
    icn[tid] = frcp(cn + EPS_NORM);
  }
  __syncthreads();
  if (tid < 64) {  // xg2 = h @ fadj2n
    int c = tid >> 4, g = tid & 15;
    float acc = 0.f;
    for (int f = 0; f < 16; ++f) acc += hact[c * 16 + f] * fn[f * 16 + g];
    xg2[tid] = acc * icn[g];
  }
  __syncthreads();
  if (tid < 32) {  // h2[d] = vec(xg2) . W2[k,d,:,:,0]
    const float* w2p = W2 + (size_t)(k * 32 + tid) * 64;
    float acc = 0.f;
    for (int i = 0; i < 64; ++i) acc += xg2[i] * w2p[i];
    h2raw[(b * 2 + k) * 32 + tid] = acc;
  }
}

// ---------------------------------------------------------------------------
// Kernel 4: BN2 stats per (k,d) over b -> 64 blocks
// ---------------------------------------------------------------------------
__global__ __launch_bounds__(256) void k4_bn2_stats(
    const float* __restrict__ h2raw, float* __restrict__ st2)
{
  __shared__ float rs[256], rq[256];
  int kd = blockIdx.x, k = kd >> 5, d = kd & 31, tid = threadIdx.x;
  float s = 0.f, q = 0.f;
  for (int bb = tid; bb < 512; bb += 256) {
    float v = h2raw[(bb * 2 + k) * 32 + d];
    s += v; q += v * v;
  }
  rs[tid] = s; rq[tid] = q; __syncthreads();
  for (int o = 128; o > 0; o >>= 1) {
    if (tid < o) { rs[tid] += rs[tid + o]; rq[tid] += rq[tid + o]; }
    __syncthreads();
  }
  if (tid == 0) {
    float m = rs[0] / 512.f;
    st2[kd * 2] = m;
    st2[kd * 2 + 1] = rq[0] / 512.f - m * m;
  }
}

// ---------------------------------------------------------------------------
// Kernel 5: BN2 + softsign + classifier (64 -> relu 32 -> 40)
// ---------------------------------------------------------------------------
__global__ __launch_bounds__(64) void k5_classify(
    const float* __restrict__ h2raw, const float* __restrict__ st2,
    const float* __restrict__ g2, const float* __restrict__ b2,
    const float* __restrict__ Wc1, const float* __restrict__ bc1,
    const float* __restrict__ Wc2, const float* __restrict__ bc2,
    float* __restrict__ out)
{
  __shared__ float flat[64], hid[32];
  int b = blockIdx.x, tid = threadIdx.x;
  {
    int k = tid >> 5, d = tid & 31;           // flat[kc], kc = k*32+d = tid
    float raw = h2raw[(b * 2 + k) * 32 + d];
    float m = st2[tid * 2], v = st2[tid * 2 + 1];
    float z = (raw - m) * frsq(v + EPS_BN) * g2[tid] + b2[tid];
    flat[tid] = softsign(z);
  }
  __syncthreads();
  if (tid < 32) {
    float acc = bc1[tid];
    for (int i = 0; i < 64; ++i) acc += flat[i] * Wc1[i * 32 + tid];
    hid[tid] = fmaxf(acc, 0.f);
  }
  __syncthreads();
  if (tid < NCLS_) {
    float acc = bc2[tid];
    for (int j = 0; j < 32; ++j) acc += hid[j] * Wc2[j * 40 + tid];
    out[b * 40 + tid] = acc;
  }
}

// ---------------------------------------------------------------------------
extern "C" void kernel_launch(void* const* d_in, const int* in_sizes, int n_in,
                              void* d_out, int out_size, void* d_ws, size_t ws_size,
                              hipStream_t stream)
{
  (void)in_sizes; (void)n_in; (void)out_size; (void)ws_size;
  const float* x   = (const float*)d_in[0];
  const float* nb  = (const float*)d_in[1];
  const float* W1  = (const float*)d_in[2];
  const float* g1  = (const float*)d_in[3];
  const float* b1  = (const float*)d_in[4];
  const float* W2  = (const float*)d_in[5];
  const float* g2  = (const float*)d_in[6];
  const float* b2  = (const float*)d_in[7];
  const float* Wc1 = (const float*)d_in[8];
  const float* bc1 = (const float*)d_in[9];
  const float* Wc2 = (const float*)d_in[10];
  const float* bc2 = (const float*)d_in[11];
  float* out = (float*)d_out;
  float* ws  = (float*)d_ws;

  float* hnb  = ws;                                    // B*K*2112 floats (8.6MB)
  float* st1  = ws + (size_t)B_ * K_ * HNB_STRIDE;     // 16
  float* h2rw = st1 + 16;                              // B*K*32
  float* st2  = h2rw + (size_t)B_ * K_ * 32;           // 128

  size_t smem1 = (size_t)(2 * JROWS * 512 + 3 * 512) * sizeof(float); // ~138KB

  k1_graph   <<<dim3(B_ * K_), dim3(256), smem1, stream>>>(x, nb, W1, hnb);
  k2_bn1_stats<<<dim3(8),      dim3(256), 0,     stream>>>(hnb, st1);
  k3_layer2  <<<dim3(B_ * K_), dim3(128), 0,     stream>>>(hnb, st1, g1, b1, W2, h2rw);
  k4_bn2_stats<<<dim3(64),     dim3(256), 0,     stream>>>(h2rw, st2);
  k5_classify<<<dim3(B_),      dim3(64),  0,     stream>>>(h2rw, st2, g2, b2,
                                                           Wc1, bc1, Wc2, bc2, out);
}